// MultiOutSizeLinear_37503654429378
// MI455X (gfx1250) — compile-verified
//
#include <hip/hip_runtime.h>

typedef __attribute__((ext_vector_type(2))) float v2f;
typedef __attribute__((ext_vector_type(4))) float v4f;
typedef __attribute__((ext_vector_type(8))) float v8f;

#define IN_FEAT   1024
#define MAX_OUT   1024
#define NBRANCH   4

#define MBLK      128              // tokens per block (8 m-tiles of 16)
#define MT        8                // m-tiles per wave
#define KSTAGE    64               // K depth staged in LDS per round
#define NSTAGE    (IN_FEAT / KSTAGE)
#define LSTR      (KSTAGE + 4)     // 68: row r starts at bank (4r)%64 -> conflict-free b64

// ---------------- kernel 1: zero the per-branch counters ----------------
__global__ void mosl_zero_cnt(unsigned* cnt) {
    if (threadIdx.x < NBRANCH) cnt[threadIdx.x] = 0u;
}

// ---------------- kernel 2: compact token ids per branch ----------------
__global__ void mosl_classify(const int* __restrict__ ofs,
                              unsigned* __restrict__ cnt,
                              int* __restrict__ lists, int ntok) {
    int t = blockIdx.x * blockDim.x + threadIdx.x;
    if (t >= ntok) return;
    int feat = ofs[t];                 // in {128,256,512,1024}
    int b = __ffs(feat) - 8;           // 128->0, 256->1, 512->2, 1024->3
    unsigned pos = atomicAdd(&cnt[b], 1u);
    lists[(size_t)b * ntok + pos] = t;
}

// ---------------- kernel 3: zero the tail [feat, 1024) of each row -------
__global__ void mosl_tail_zero(const int* __restrict__ ofs,
                               float* __restrict__ out) {
    int t = blockIdx.x;
    int feat = ofs[t];
    int c = threadIdx.x * 4;           // 256 threads cover 1024 floats as v4f
    if (c >= feat) {
        v4f z = {};
        *(v4f*)(out + (size_t)t * MAX_OUT + c) = z;
    }
}

// ---------------- kernel 4: branch-compacted WMMA GEMM ------------------
// block = 256 threads = 8 wave32; block tile = 128 tokens x 128 cols.
// Each wave owns one 16-col slice and 8 independent 16x16 accumulators,
// so one B fetch feeds 8 pipelined v_wmma_f32_16x16x4_f32 ops.
__global__ void __launch_bounds__(256)
mosl_gemm(const float* __restrict__ x,
          const float* __restrict__ w,
          const float* __restrict__ bias,
          const int* __restrict__ lists,
          const unsigned* __restrict__ cnt,
          float* __restrict__ out, int ntok) {
    const int b = blockIdx.z;
    const int nblocks128 = 1 << b;              // N_b/128 : 1,2,4,8
    if ((int)blockIdx.y >= nblocks128) return;
    const unsigned count = cnt[b];
    const unsigned m_base = blockIdx.x * (unsigned)MBLK;
    if (m_base >= count) return;

    __shared__ __align__(16) float sA[MBLK * LSTR];   // 34816 B
    __shared__ int sTok[MBLK];

    const int tid  = threadIdx.x;
    const int lane = tid & 31;
    const int wave = tid >> 5;
    const int half = lane >> 4;                 // 0: K={0,1}, 1: K={2,3}
    const int ml   = lane & 15;                 // M (A) / N (B,D) index

    // gather token ids; pad ragged tile with first (valid) token, stores masked
    if (tid < MBLK) {
        unsigned mi = m_base + tid;
        sTok[tid] = (mi < count) ? lists[(size_t)b * ntok + mi]
                                 : lists[(size_t)b * ntok + m_base];
    }
    __syncthreads();

    const int n0 = blockIdx.y * 128 + wave * 16;
    const float* wB = w + (size_t)b * (MAX_OUT * IN_FEAT)
                        + (size_t)(n0 + ml) * IN_FEAT + 2 * half;

    v8f c[MT];
    #pragma unroll
    for (int mt = 0; mt < MT; ++mt) c[mt] = (v8f){};

    for (int stage = 0; stage < NSTAGE; ++stage) {
        // stage 128 rows x 64 K of gathered x into LDS (v4f per lane)
        #pragma unroll
        for (int i = 0; i < 8; ++i) {
            int f    = tid + i * 256;           // 0..2047 v4f slots
            int row  = f >> 4;                  // 16 v4f per row
            int col4 = (f & 15) << 2;
            v4f v = *(const v4f*)(x + (size_t)sTok[row] * IN_FEAT
                                    + stage * KSTAGE + col4);
            *(v4f*)&sA[row * LSTR + col4] = v;
        }
        __syncthreads();

        const float* lB = wB + stage * KSTAGE;
        const float* lA = &sA[ml * LSTR + 2 * half];
        #pragma unroll 4
        for (int kk = 0; kk < KSTAGE; kk += 4) {
            v2f bv = *(const v2f*)(lB + kk);    // B: w[n0+ml, k+2h .. +1]
            #pragma unroll
            for (int mt = 0; mt < MT; ++mt) {   // 8 independent accumulators
                v2f a = *(const v2f*)(lA + mt * (16 * LSTR) + kk);
                c[mt] = __builtin_amdgcn_wmma_f32_16x16x4_f32(
                            false, a, false, bv, (short)0, c[mt], false, false);
            }
        }
        __syncthreads();                        // protect sA before next fill
    }

    // bias: every D element in this lane has column n0+ml
    const float bs = bias[(size_t)b * MAX_OUT + n0 + ml];

    // D layout: VGPR r -> row (r + 8*half), col ml ; mask ragged rows
    #pragma unroll
    for (int mt = 0; mt < MT; ++mt) {
        #pragma unroll
        for (int r = 0; r < 8; ++r) {
            int mrow = mt * 16 + r + 8 * half;
            if (m_base + (unsigned)mrow < count) {
                out[(size_t)sTok[mrow] * MAX_OUT + n0 + ml] = c[mt][r] + bs;
            }
        }
    }
}

extern "C" void kernel_launch(void* const* d_in, const int* in_sizes, int n_in,
                              void* d_out, int out_size, void* d_ws, size_t ws_size,
                              hipStream_t stream) {
    const float* x    = (const float*)d_in[0];  // [32,2048,1024]
    const float* w    = (const float*)d_in[1];  // [4,1024,1024]
    const float* bias = (const float*)d_in[2];  // [4,1024]
    const int*   ofs  = (const int*)d_in[3];    // [32,2048]
    float* out = (float*)d_out;

    const int ntok = in_sizes[3];               // 65536

    unsigned* cnt = (unsigned*)d_ws;
    int* lists = (int*)((char*)d_ws + 256);     // 4 * ntok ints

    mosl_zero_cnt<<<1, 64, 0, stream>>>(cnt);
    mosl_classify<<<(ntok + 255) / 256, 256, 0, stream>>>(ofs, cnt, lists, ntok);
    mosl_tail_zero<<<ntok, 256, 0, stream>>>(ofs, out);

    dim3 grid((ntok + MBLK - 1) / MBLK, 8, NBRANCH);
    mosl_gemm<<<grid, 256, 0, stream>>>(x, w, bias, lists, cnt, out, ntok);
}